// GATLayer_80633716015136
// MI455X (gfx1250) — compile-verified
//
#include <hip/hip_runtime.h>

// ---------------------------------------------------------------------------
// GAT layer for MI455X (gfx1250, wave32).
//   feature[N,256], W_lin[256,256], W_attn[256,256], a[4,128], src[E], dst[E]
//   out[N,256]
// GEMMs run on V_WMMA_F32_16X16X4_F32 (fp32 in/out: matches reference math).
// Feature tiles staged to LDS via GLOBAL_LOAD_ASYNC_TO_LDS_B128 (ASYNCcnt).
// Edge softmax + scatter use hardware f32 atomics (L2-resident working set).
// ---------------------------------------------------------------------------

typedef float v2f __attribute__((ext_vector_type(2)));
typedef float v8f __attribute__((ext_vector_type(8)));

#define GAT_IN   256
#define GAT_OUT  256
#define GAT_H    4
#define GAT_D    64
// 260 % 64 == 4 -> LDS bank = 4*row + k: conflict-free for 16 rows, and every
// row base (260*4 = 1040 B) stays 16-B aligned for async B128 stores.
#define LDS_STRIDE 260

// --------------------------------------------------------------------------
// Kernel 1: fused dual GEMM + attention logits.
// One block = 16 nodes. 8 waves: waves 0-3 -> key GEMM head h=wave (folds the
// a-dot into alpha_src/alpha_dst via in-wave shuffle reduction, key is never
// materialized); waves 4-7 -> val GEMM head h=wave-4, stored to workspace.
// --------------------------------------------------------------------------
__global__ __launch_bounds__(256)
void gat_gemm_kernel(const float* __restrict__ feature,
                     const float* __restrict__ W_lin,
                     const float* __restrict__ W_attn,
                     const float* __restrict__ a,
                     float* __restrict__ val,
                     float* __restrict__ alpha_src,
                     float* __restrict__ alpha_dst,
                     int N)
{
    __shared__ float fa[16 * LDS_STRIDE];

    const int tid       = threadIdx.x;
    const int m0        = blockIdx.x * 16;
    const bool tile_full = (m0 + 16 <= N);        // uniform across block

    // Stage 16x256 fp32 feature tile into LDS with async B128 copies
    // (4 x 16B per thread), tracked on ASYNCcnt -- no VGPR round trip.
    for (int i = tid; i < 16 * (GAT_IN / 4); i += 256) {
        const int r  = i / (GAT_IN / 4);
        const int c4 = i % (GAT_IN / 4);
        const int rr = (m0 + r < N) ? (m0 + r) : (N - 1);
        const float* gaddr = feature + (size_t)rr * GAT_IN + c4 * 4;
        const unsigned lds_off =
            (unsigned)(uintptr_t)(&fa[r * LDS_STRIDE + c4 * 4]);
        asm volatile("global_load_async_to_lds_b128 %0, %1, off"
                     :: "v"(lds_off), "v"(gaddr)
                     : "memory");
    }
    asm volatile("s_wait_asynccnt 0x0" ::: "memory");
    __syncthreads();

    const int  wave    = tid >> 5;
    const int  lane    = tid & 31;
    const int  half    = lane >> 4;     // K-half select per ISA A/B layout
    const int  lm      = lane & 15;     // m (A) / n (B) index
    const bool is_attn = (wave < 4);
    const int  h       = is_attn ? wave : (wave - 4);
    const float* __restrict__ W = is_attn ? W_attn : W_lin;

    float asrc[8], adst[8];
#pragma unroll
    for (int r = 0; r < 8; ++r) { asrc[r] = 0.f; adst[r] = 0.f; }

    const int koff = half * 2;

    for (int t = 0; t < 4; ++t) {                 // 4 x 16 cols = one head (64)
        const int c0 = h * GAT_D + t * 16;
        v8f acc = {0.f, 0.f, 0.f, 0.f, 0.f, 0.f, 0.f, 0.f};
        const float* __restrict__ arow = &fa[lm * LDS_STRIDE];
        const float* __restrict__ wrow = &W[(size_t)(c0 + lm) * GAT_IN];

#pragma unroll 4
        for (int kk = 0; kk < GAT_IN; kk += 4) {
            // A: 16x4 fp32 fragment, lane holds (m=lm, K=koff..koff+1)
            const v2f A = *(const v2f*)&arow[kk + koff];
            // B: 4x16 fp32 fragment = W^T, lane holds (n=lm, K=koff..koff+1)
            const v2f B = *(const v2f*)&wrow[kk + koff];
            acc = __builtin_amdgcn_wmma_f32_16x16x4_f32(
                false, A, false, B, (short)0, acc, false, false);
        }

        if (!is_attn) {
            // C layout: acc[r] = element (row m0 + r + half*8, col c0 + lm)
            if (tile_full) {
#pragma unroll
                for (int r = 0; r < 8; ++r)
                    val[(size_t)(m0 + r + half * 8) * GAT_OUT + c0 + lm] = acc[r];
            } else {
                for (int r = 0; r < 8; ++r) {
                    const int row = m0 + r + half * 8;
                    if (row < N) val[(size_t)row * GAT_OUT + c0 + lm] = acc[r];
                }
            }
        } else {
            const float av = a[h * (2 * GAT_D) + (t * 16 + lm)];           // a[h, d]
            const float ad = a[h * (2 * GAT_D) + GAT_D + (t * 16 + lm)];   // a[h, D+d]
#pragma unroll
            for (int r = 0; r < 8; ++r) {
                asrc[r] += acc[r] * av;
                adst[r] += acc[r] * ad;
            }
        }
    }

    if (is_attn) {
        // Reduce over the 16 columns held by each 16-lane half (xor 8/4/2/1
        // stays inside a half on wave32).
#pragma unroll
        for (int r = 0; r < 8; ++r) {
#pragma unroll
            for (int off = 8; off; off >>= 1) {
                asrc[r] += __shfl_xor(asrc[r], off, 32);
                adst[r] += __shfl_xor(adst[r], off, 32);
            }
        }
        if (lm == 0) {
            if (tile_full) {
#pragma unroll
                for (int r = 0; r < 8; ++r) {
                    const int row = m0 + r + half * 8;
                    alpha_src[(size_t)row * GAT_H + h] = asrc[r];
                    alpha_dst[(size_t)row * GAT_H + h] = adst[r];
                }
            } else {
                for (int r = 0; r < 8; ++r) {
                    const int row = m0 + r + half * 8;
                    if (row < N) {
                        alpha_src[(size_t)row * GAT_H + h] = asrc[r];
                        alpha_dst[(size_t)row * GAT_H + h] = adst[r];
                    }
                }
            }
        }
    }
}

// --------------------------------------------------------------------------
// Kernel 0: init smax = -inf, denom = 0, out = 0 (harness poisons buffers).
// --------------------------------------------------------------------------
__global__ void gat_init_kernel(float* __restrict__ smax,
                                float* __restrict__ denom,
                                float* __restrict__ out,
                                int nh, int nout)
{
    const int g = blockIdx.x * blockDim.x + threadIdx.x;
    if (g < nh) {
        smax[g]  = -__builtin_inff();
        denom[g] = 0.f;
    }
    if (g < nout) out[g] = 0.f;
}

__device__ __forceinline__ float leaky_relu02(float x) {
    return x > 0.f ? x : 0.2f * x;
}

// --------------------------------------------------------------------------
// Kernel 2: per-edge score -> segment max over dst (hw f32 atomic max).
// --------------------------------------------------------------------------
__global__ __launch_bounds__(256)
void gat_edge_max_kernel(const int* __restrict__ src,
                         const int* __restrict__ dst,
                         const float* __restrict__ alpha_src,
                         const float* __restrict__ alpha_dst,
                         float* __restrict__ smax,
                         int E)
{
    const int e = blockIdx.x * blockDim.x + threadIdx.x;
    if (e >= E) return;
    const int s = src[e];
    const int d = dst[e];
#pragma unroll
    for (int h = 0; h < GAT_H; ++h) {
        const float sc = leaky_relu02(alpha_src[(size_t)s * GAT_H + h] +
                                      alpha_dst[(size_t)d * GAT_H + h]);
        __hip_atomic_fetch_max(&smax[(size_t)d * GAT_H + h], sc,
                               __ATOMIC_RELAXED, __HIP_MEMORY_SCOPE_AGENT);
    }
}

// --------------------------------------------------------------------------
// Kernel 3: ex = exp(score - smax[dst]); denom[dst] += ex (hw f32 atomic add).
// --------------------------------------------------------------------------
__global__ __launch_bounds__(256)
void gat_edge_expsum_kernel(const int* __restrict__ src,
                            const int* __restrict__ dst,
                            const float* __restrict__ alpha_src,
                            const float* __restrict__ alpha_dst,
                            const float* __restrict__ smax,
                            float* __restrict__ denom,
                            float* __restrict__ ex,
                            int E)
{
    const int e = blockIdx.x * blockDim.x + threadIdx.x;
    if (e >= E) return;
    const int s = src[e];
    const int d = dst[e];
#pragma unroll
    for (int h = 0; h < GAT_H; ++h) {
        const float sc = leaky_relu02(alpha_src[(size_t)s * GAT_H + h] +
                                      alpha_dst[(size_t)d * GAT_H + h]);
        const float x = __expf(sc - smax[(size_t)d * GAT_H + h]);
        ex[(size_t)e * GAT_H + h] = x;
        __hip_atomic_fetch_add(&denom[(size_t)d * GAT_H + h], x,
                               __ATOMIC_RELAXED, __HIP_MEMORY_SCOPE_AGENT);
    }
}

// --------------------------------------------------------------------------
// Kernel 4: out[dst] += (ex/denom[dst]) * val[src]. 64 threads per edge so
// val reads and out atomics are coalesced 256B bursts (L2-resident).
// --------------------------------------------------------------------------
__global__ __launch_bounds__(256)
void gat_edge_scatter_kernel(const int* __restrict__ src,
                             const int* __restrict__ dst,
                             const float* __restrict__ ex,
                             const float* __restrict__ denom,
                             const float* __restrict__ val,
                             float* __restrict__ out,
                             int E)
{
    const long long g = (long long)blockIdx.x * blockDim.x + threadIdx.x;
    const int e  = (int)(g >> 6);
    const int dd = (int)(g & 63);
    if (e >= E) return;
    const int s = src[e];
    const int d = dst[e];
#pragma unroll
    for (int h = 0; h < GAT_H; ++h) {
        const float w = ex[(size_t)e * GAT_H + h] /
                        denom[(size_t)d * GAT_H + h];
        const float v = val[(size_t)s * GAT_OUT + h * GAT_D + dd];
        __hip_atomic_fetch_add(&out[(size_t)d * GAT_OUT + h * GAT_D + dd],
                               w * v, __ATOMIC_RELAXED,
                               __HIP_MEMORY_SCOPE_AGENT);
    }
}

// --------------------------------------------------------------------------
extern "C" void kernel_launch(void* const* d_in, const int* in_sizes, int n_in,
                              void* d_out, int out_size, void* d_ws, size_t ws_size,
                              hipStream_t stream)
{
    const float* feature = (const float*)d_in[0];   // [N, 256]
    const float* W_lin   = (const float*)d_in[1];   // [256, 256]
    const float* W_attn  = (const float*)d_in[2];   // [256, 256]
    const float* a       = (const float*)d_in[3];   // [4, 128]
    const int*   src     = (const int*)d_in[4];     // [E]
    const int*   dst     = (const int*)d_in[5];     // [E]
    float*       out     = (float*)d_out;           // [N, 256]

    const int N = in_sizes[0] / GAT_IN;
    const int E = in_sizes[4];

    // Workspace layout (fp32): val | alpha_src | alpha_dst | smax | denom | ex
    float* val       = (float*)d_ws;                         // N*256
    float* alpha_src = val       + (size_t)N * GAT_OUT;      // N*H
    float* alpha_dst = alpha_src + (size_t)N * GAT_H;        // N*H
    float* smax      = alpha_dst + (size_t)N * GAT_H;        // N*H
    float* denom     = smax      + (size_t)N * GAT_H;        // N*H
    float* ex        = denom     + (size_t)N * GAT_H;        // E*H

    const int nh   = N * GAT_H;
    const int nout = N * GAT_OUT;

    gat_init_kernel<<<(nout + 255) / 256, 256, 0, stream>>>(smax, denom, out, nh, nout);

    gat_gemm_kernel<<<(N + 15) / 16, 256, 0, stream>>>(
        feature, W_lin, W_attn, a, val, alpha_src, alpha_dst, N);

    gat_edge_max_kernel<<<(E + 255) / 256, 256, 0, stream>>>(
        src, dst, alpha_src, alpha_dst, smax, E);

    gat_edge_expsum_kernel<<<(E + 255) / 256, 256, 0, stream>>>(
        src, dst, alpha_src, alpha_dst, smax, denom, ex, E);

    const long long sc_threads = (long long)E * GAT_D;
    gat_edge_scatter_kernel<<<(unsigned)((sc_threads + 255) / 256), 256, 0, stream>>>(
        src, dst, ex, denom, val, out, E);
}